// Anchor_Assigner_67765993996564
// MI455X (gfx1250) — compile-verified
//
#include <hip/hip_runtime.h>
#include <stdint.h>

#define BB 8
#define TT 64
#define FF 84
#define NN 100000

typedef float        f32x4 __attribute__((ext_vector_type(4)));
typedef unsigned int u32x4 __attribute__((ext_vector_type(4)));
typedef int          i32x4 __attribute__((ext_vector_type(4)));
typedef int          i32x8 __attribute__((ext_vector_type(8)));

// iou >= 0, so float bits are monotonic; ~a makes ties pick the LOWEST anchor
// index under max(), matching jnp.argmax's first-hit semantics.
__device__ __forceinline__ unsigned long long pack_key(float iou, unsigned a) {
    return ((unsigned long long)__float_as_uint(iou) << 32) |
           (unsigned long long)(0xFFFFFFFFu - a);
}

__global__ void k_init_keys(unsigned long long* __restrict__ keys) {
    int i = blockIdx.x * blockDim.x + threadIdx.x;
    if (i < BB * TT) keys[i] = 0ull;
}

__global__ __launch_bounds__(256)
void k_iou(const float* __restrict__ labels,
           const float* __restrict__ anchors,
           float* __restrict__ maxiou,
           int* __restrict__ tgt,
           unsigned long long* __restrict__ gkeys) {
    __shared__ __align__(16) float s_tile[TT * 4];   // TDM destination: cxcywh per target
    __shared__ float s_box[TT][4];                   // xyxy
    __shared__ float s_area[TT];
    __shared__ unsigned long long s_keys[TT];

    const int b = blockIdx.y;
    const int a = blockIdx.x * blockDim.x + threadIdx.x;

    // ---- Stage labels[b][:, 0:4] into LDS with the Tensor Data Mover ----
    // 2-D tile: tile_dim0=4 elems of a tensor_dim0_stride=84 row, 64 rows,
    // packed contiguously into LDS (64 x 16B). Issued once by wave 0.
    if (threadIdx.x < 32u) {
#if __has_builtin(__builtin_amdgcn_tensor_load_to_lds)
        const uint64_t gaddr = (uint64_t)(uintptr_t)(labels + (size_t)b * TT * FF);
        const uint32_t laddr = (uint32_t)(uintptr_t)(&s_tile[0]);
        u32x4 g0;
        g0.x = 1u;                                            // count=1, is_restore=0, gather off
        g0.y = laddr;                                         // lds_addr       [63:32]
        g0.z = (uint32_t)(gaddr & 0xFFFFFFFFull);             // global_addr    [95:64]
        g0.w = (uint32_t)((gaddr >> 32) & 0x01FFFFFFull)      // global_addr    [120:96]
             | (2u << 30);                                    // type=2 (image) [127:126]
        i32x8 g1;
        g1[0] = (int)(2u << 16);                // workgroup_mask=0, data_size=2 (4 bytes)
        g1[1] = (int)(84u << 16);               // tensor_dim0[15:0] @ bits 63:48
        g1[2] = (int)(64u << 16);               // tensor_dim0 hi=0, tensor_dim1[15:0]=64
        g1[3] = (int)(4u << 16);                // tensor_dim1 hi=0, tile_dim0=4
        g1[4] = (int)(64u);                     // tile_dim1=64, tile_dim2=0
        g1[5] = (int)(84u);                     // tensor_dim0_stride low32 = 84
        g1[6] = 0;                              // stride0 hi, tensor_dim1_stride low
        g1[7] = 0;
        i32x4 gz4 = {0, 0, 0, 0};               // groups 2/3 unused (2-D tensor)
        i32x8 gz8 = {0, 0, 0, 0, 0, 0, 0, 0};   // extra group (clang-23 6-arg form)
        __builtin_amdgcn_tensor_load_to_lds(g0, g1, gz4, gz4, gz8, 0);
        __builtin_amdgcn_s_wait_tensorcnt(0);
#else
        for (int t = threadIdx.x; t < TT * 4; t += 32)
            s_tile[t] = labels[(size_t)b * TT * FF + (size_t)(t >> 2) * FF + (t & 3)];
#endif
    }
    if (threadIdx.x < TT) s_keys[threadIdx.x] = 0ull;
    __syncthreads();

    // cxcywh -> xyxy + area, one target per lane of the first two waves
    if (threadIdx.x < TT) {
        const int t = threadIdx.x;
        float cx = s_tile[t * 4 + 0], cy = s_tile[t * 4 + 1];
        float w  = s_tile[t * 4 + 2], h  = s_tile[t * 4 + 3];
        float x0 = cx - 0.5f * w, y0 = cy - 0.5f * h;
        float x1 = cx + 0.5f * w, y1 = cy + 0.5f * h;
        s_box[t][0] = x0; s_box[t][1] = y0; s_box[t][2] = x1; s_box[t][3] = y1;
        s_area[t] = (x1 - x0) * (y1 - y0);
    }
    __syncthreads();

    if (a < NN) {
        f32x4 an = ((const f32x4*)anchors)[a];
        float ax0 = an.x - 0.5f * an.z, ay0 = an.y - 0.5f * an.w;
        float ax1 = an.x + 0.5f * an.z, ay1 = an.y + 0.5f * an.w;
        float area_a = (ax1 - ax0) * (ay1 - ay0);

        float best = -1.0f;
        int bestT = 0;
        for (int t = 0; t < TT; ++t) {
            float ltx = fmaxf(ax0, s_box[t][0]);
            float lty = fmaxf(ay0, s_box[t][1]);
            float rbx = fminf(ax1, s_box[t][2]);
            float rby = fminf(ay1, s_box[t][3]);
            float iw = fmaxf(rbx - ltx, 0.0f);
            float ih = fmaxf(rby - lty, 0.0f);
            float inter = iw * ih;
            float iou = inter / (area_a + s_area[t] - inter + 1e-9f);
            if (iou > best) { best = iou; bestT = t; }   // strict '>' => first max wins
            // cross-anchor argmax for this target: LDS reduction first
            unsigned long long k = pack_key(iou, (unsigned)a);
            if (k > s_keys[t]) atomicMax(&s_keys[t], k); // value only grows: filter is safe
        }
        maxiou[(size_t)b * NN + a] = best;
        tgt[(size_t)b * NN + a]   = bestT;
    }
    __syncthreads();

    // one global atomic per (block, target)
    if (threadIdx.x < TT)
        atomicMax(&gkeys[b * TT + threadIdx.x], s_keys[threadIdx.x]);
}

__global__ void k_masks(const float* __restrict__ maxiou,
                        float* __restrict__ fore, float* __restrict__ back) {
    size_t i = (size_t)blockIdx.x * blockDim.x + threadIdx.x;
    if (i < (size_t)BB * NN) {
        float m = maxiou[i];
        float f = (m >= 0.5f) ? 1.0f : 0.0f;
        float bk = (f == 0.0f && m < 0.4f) ? 1.0f : 0.0f;
        fore[i] = f;
        back[i] = bk;
    }
}

__global__ void k_fmt(const unsigned long long* __restrict__ gkeys,
                      float* __restrict__ fore, float* __restrict__ back) {
    int i = blockIdx.x * blockDim.x + threadIdx.x;
    if (i < BB * TT) {
        int b = i / TT;
        unsigned a = 0xFFFFFFFFu - (unsigned)(gkeys[i] & 0xFFFFFFFFull);
        if (a < NN) {
            fore[(size_t)b * NN + a] = 1.0f;   // fm_t override
            back[(size_t)b * NN + a] = 0.0f;   // back = !fore & (...)
        }
    }
}

// assigned_target gather: one thread per 16B of output. Label rows are
// 84 floats = 336B = 21 float4 -> fully float4-aligned. NT stores keep the
// 268MB stream out of L2 so labels/tgt stay resident.
__global__ __launch_bounds__(256)
void k_copy(const float* __restrict__ labels,
            const int* __restrict__ tgt,
            float* __restrict__ out) {
    size_t idx = (size_t)blockIdx.x * blockDim.x + threadIdx.x;
    const size_t total = (size_t)BB * NN * 21;
    if (idx >= total) return;
    int c = (int)(idx % 21);
    size_t ba = idx / 21;                  // b*N + a
    int b = (int)(ba / NN);
    int t = tgt[ba];                       // shared by 21 consecutive threads (cached)
    const f32x4* lrow = (const f32x4*)(labels + ((size_t)b * TT + t) * FF);
    f32x4 v = lrow[c];
    __builtin_nontemporal_store(v, ((f32x4*)out) + ba * 21 + c);
}

extern "C" void kernel_launch(void* const* d_in, const int* in_sizes, int n_in,
                              void* d_out, int out_size, void* d_ws, size_t ws_size,
                              hipStream_t stream) {
    const float* labels  = (const float*)d_in[0];   // (8,64,84) f32
    const float* anchors = (const float*)d_in[1];   // (1,100000,4) f32

    float* fore     = (float*)d_out;                       // (8,100000)
    float* back     = fore + (size_t)BB * NN;              // (8,100000)
    float* assigned = back + (size_t)BB * NN;              // (8,100000,84)

    char* ws = (char*)d_ws;
    float* maxiou            = (float*)ws;                                   // 3.2 MB
    int*   tgt               = (int*)(ws + (size_t)BB * NN * sizeof(float)); // 3.2 MB
    unsigned long long* keys = (unsigned long long*)(ws + 2 * (size_t)BB * NN * sizeof(float)); // 4 KB

    k_init_keys<<<2, 256, 0, stream>>>(keys);

    dim3 g2((NN + 255) / 256, BB);
    k_iou<<<g2, 256, 0, stream>>>(labels, anchors, maxiou, tgt, keys);

    k_masks<<<(BB * NN + 255) / 256, 256, 0, stream>>>(maxiou, fore, back);
    k_fmt<<<2, 256, 0, stream>>>(keys, fore, back);

    size_t totc = (size_t)BB * NN * 21;
    k_copy<<<(unsigned)((totc + 255) / 256), 256, 0, stream>>>(labels, tgt, assigned);
}